// Spardmax_7902739825000
// MI455X (gfx1250) — compile-verified
//
#include <hip/hip_runtime.h>

#define COLS 4096
#define TPB  256      // 8 wave32s
// each thread owns 16 elements = 4 x float4

__global__ __launch_bounds__(TPB) void sparsemax_st_kernel(
    const float* __restrict__ x, float* __restrict__ y) {
  __shared__ float redS[8];
  __shared__ float redC[8];
  __shared__ float bc[2];                 // [0]=tau, [1]=done flag
  extern __shared__ float smem[];         // 4096 floats: staged row

  const int tid  = threadIdx.x;
  const int row  = blockIdx.x;
  const int lane = tid & 31;
  const int wid  = tid >> 5;

  // ---- CDNA5 async global->LDS stage of one row (ASYNCcnt path) ----
  // per lane: LDS addr = dyn base + tid*16 + offset; global addr = row base + tid*16 + offset
  unsigned lds_dyn  = __builtin_amdgcn_groupstaticsize();   // dynamic LDS starts after statics
  unsigned lds_addr = lds_dyn + (unsigned)tid * 16u;
  unsigned goff     = (unsigned)tid * 16u;
  unsigned long long gbase =
      (unsigned long long)(const void*)(x + (size_t)row * COLS);
  asm volatile(
      "global_load_async_to_lds_b128 %0, %1, %2 offset:0\n\t"
      "global_load_async_to_lds_b128 %0, %1, %2 offset:4096\n\t"
      "global_load_async_to_lds_b128 %0, %1, %2 offset:8192\n\t"
      "global_load_async_to_lds_b128 %0, %1, %2 offset:12288"
      :
      : "v"(lds_addr), "v"(goff), "s"(gbase)
      : "memory");
  asm volatile("s_wait_asynccnt 0x0" ::: "memory");
  __syncthreads();

  // ---- pull this thread's 16 elements into registers; find row max ----
  const float4* sm4 = reinterpret_cast<const float4*>(smem);
  float4 rv[4];
  float lmax = -3.402823466e38f;
#pragma unroll
  for (int j = 0; j < 4; ++j) {
    rv[j] = sm4[tid + j * 256];
    lmax  = fmaxf(lmax, fmaxf(fmaxf(rv[j].x, rv[j].y), fmaxf(rv[j].z, rv[j].w)));
  }
#pragma unroll
  for (int o = 16; o > 0; o >>= 1)
    lmax = fmaxf(lmax, __shfl_down(lmax, o, 32));
  if (lane == 0) redS[wid] = lmax;
  __syncthreads();
  if (tid == 0) {
    float m = redS[0];
#pragma unroll
    for (int w = 1; w < 8; ++w) m = fmaxf(m, redS[w]);
    bc[0] = m - 1.0f;   // tau0: f(tau0) >= 0 guaranteed
    bc[1] = 0.0f;
  }
  __syncthreads();
  float tau = bc[0];

  // ---- Newton on f(tau) = sum(relu(x - tau)) - 1  (convex piecewise-linear) ----
  for (int it = 0; it < 32; ++it) {
    float s = 0.0f, c = 0.0f;
#pragma unroll
    for (int j = 0; j < 4; ++j) {
      float4 v = rv[j];
      if (v.x > tau) { s += v.x - tau; c += 1.0f; }
      if (v.y > tau) { s += v.y - tau; c += 1.0f; }
      if (v.z > tau) { s += v.z - tau; c += 1.0f; }
      if (v.w > tau) { s += v.w - tau; c += 1.0f; }
    }
#pragma unroll
    for (int o = 16; o > 0; o >>= 1) {
      s += __shfl_down(s, o, 32);
      c += __shfl_down(c, o, 32);
    }
    if (lane == 0) { redS[wid] = s; redC[wid] = c; }
    __syncthreads();
    if (tid == 0) {
      float S = 0.0f, C = 0.0f;
#pragma unroll
      for (int w = 0; w < 8; ++w) { S += redS[w]; C += redC[w]; }
      float t_new = (C >= 1.0f) ? (tau + (S - 1.0f) / C) : tau;
      bc[1] = (t_new <= tau) ? 1.0f : 0.0f;  // converged (monotone ascent stalled)
      bc[0] = t_new;
    }
    __syncthreads();
    float t2   = bc[0];
    float done = bc[1];
    tau = t2;
    if (done != 0.0f) break;   // uniform across the workgroup
  }

  // ---- write straight-through forward value: support indicator ----
  float4* out4 = reinterpret_cast<float4*>(y) + (size_t)row * (COLS / 4);
#pragma unroll
  for (int j = 0; j < 4; ++j) {
    float4 v = rv[j];
    float4 o;
    o.x = (v.x > tau) ? 1.0f : 0.0f;
    o.y = (v.y > tau) ? 1.0f : 0.0f;
    o.z = (v.z > tau) ? 1.0f : 0.0f;
    o.w = (v.w > tau) ? 1.0f : 0.0f;
    out4[tid + j * 256] = o;
  }
}

extern "C" void kernel_launch(void* const* d_in, const int* in_sizes, int n_in,
                              void* d_out, int out_size, void* d_ws, size_t ws_size,
                              hipStream_t stream) {
  (void)n_in; (void)d_ws; (void)ws_size; (void)out_size;
  const float* x = (const float*)d_in[0];
  float*       y = (float*)d_out;
  const int rows = in_sizes[0] / COLS;   // 8192 for the reference shapes
  sparsemax_st_kernel<<<rows, TPB, COLS * sizeof(float), stream>>>(x, y);
}